// RandomPrompter_64982855189232
// MI455X (gfx1250) — compile-verified
//
#include <hip/hip_runtime.h>
#include <hip/hip_bf16.h>

#define ISIZE 224
#define PSIZE 30
#define NCH   3
#define CH_ELEMS  (ISIZE * ISIZE)        // 50176
#define IMG_ELEMS (NCH * CH_ELEMS)       // 150528
#define TPB 256
#define F   3                            // float4 per thread
#define BLK_ELEMS (TPB * 4 * F)          // 3072 elements per block
#define BLKS_PER_IMG (IMG_ELEMS / BLK_ELEMS) // 49

#define __AS1 __attribute__((address_space(1)))
#define __AS3 __attribute__((address_space(3)))

// gfx12-family cache policy: TH in bits [2:0]; TH_NT == 1 (non-temporal)
#define CPOL_RT 0
#define CPOL_NT 1

typedef int v4i __attribute__((ext_vector_type(4)));

#if __has_builtin(__builtin_amdgcn_global_load_async_to_lds_b128) && \
    __has_builtin(__builtin_amdgcn_global_store_async_from_lds_b128)
#define HAVE_ASYNC_LDS 1
#else
#define HAVE_ASYNC_LDS 0
#warning "gfx1250 async-to-LDS builtins NOT available on this toolchain; using VGPR copy fallback"
#endif

#if __has_builtin(__builtin_amdgcn_s_wait_asynccnt)
#define WAIT_ASYNC0() __builtin_amdgcn_s_wait_asynccnt(0)
#else
#define WAIT_ASYNC0() __builtin_amdgcn_s_waitcnt(0)
#warning "s_wait_asynccnt builtin NOT available"
#endif

__global__ __launch_bounds__(TPB) void RandomPrompter_kernel(
    const float* __restrict__ x,
    const float* __restrict__ patch,
    const int*   __restrict__ offsets,
    float*       __restrict__ out)
{
    const int img  = blockIdx.y;
    const int tid  = threadIdx.x;
    const long long imgBase = (long long)img * IMG_ELEMS;
    const int ebase = blockIdx.x * BLK_ELEMS;   // element offset within the image

    // Per-sample patch offsets; force scalar so the fast/slow branch is SALU-uniform.
    const int r = __builtin_amdgcn_readfirstlane(offsets[2 * img]);
    const int c = __builtin_amdgcn_readfirstlane(offsets[2 * img + 1]);

    // Conservative block-uniform intersection test: does this block's flat element
    // range touch any channel's patch bounding range?
    bool hit = false;
    const int p_lo_row = r * ISIZE + c;                       // first patch elem (flat, in-channel)
    const int p_hi_row = (r + PSIZE - 1) * ISIZE + c + PSIZE; // one past last
#pragma unroll
    for (int ch = 0; ch < NCH; ++ch) {
        const int lo = ch * CH_ELEMS + p_lo_row;
        const int hi = ch * CH_ELEMS + p_hi_row;
        if (ebase < hi && (ebase + BLK_ELEMS) > lo) hit = true;
    }

    if (!hit) {
        // ---- Fast path: pure streaming copy via the CDNA5 async LDS data mover ----
        // Loads RT (let x allocate in the 192MB L2 for replay reuse);
        // stores NT (write-once output must not evict x).
#if HAVE_ASYNC_LDS
        __shared__ float lds[TPB * 4 * F];
        const float* gsrc = x   + imgBase + ebase;
        float*       gdst = out + imgBase + ebase;
#pragma unroll
        for (int f = 0; f < F; ++f) {
            const int v = f * TPB + tid;             // float4 slot in block (coalesced)
            __builtin_amdgcn_global_load_async_to_lds_b128(
                (__AS1 v4i*)(gsrc + v * 4),
                (__AS3 v4i*)(&lds[v * 4]), 0, CPOL_RT);
        }
        WAIT_ASYNC0();                               // LDS now holds this wave's data
#pragma unroll
        for (int f = 0; f < F; ++f) {
            const int v = f * TPB + tid;
            __builtin_amdgcn_global_store_async_from_lds_b128(
                (__AS1 v4i*)(gdst + v * 4),
                (__AS3 v4i*)(&lds[v * 4]), 0, CPOL_NT);
        }
        // S_ENDPGM performs an implicit wait-idle; stores drain before wave retire.
#else
#pragma unroll
        for (int f = 0; f < F; ++f) {
            const int e = ebase + (f * TPB + tid) * 4;
            const float4 v = *(const float4*)(x + imgBase + e);
            float* o = out + imgBase + e;
            __builtin_nontemporal_store(v.x, o + 0);
            __builtin_nontemporal_store(v.y, o + 1);
            __builtin_nontemporal_store(v.z, o + 2);
            __builtin_nontemporal_store(v.w, o + 3);
        }
#endif
    } else {
        // ---- Slow path: copy + per-element patch add (only ~15-20% of blocks) ----
#pragma unroll
        for (int f = 0; f < F; ++f) {
            const int e   = ebase + (f * TPB + tid) * 4;
            float4 v = *(const float4*)(x + imgBase + e);
            const int ch  = e / CH_ELEMS;
            const int rem = e - ch * CH_ELEMS;
            const int row = rem / ISIZE;
            const int col = rem - row * ISIZE;
            const int pr  = row - r;
            if ((unsigned)pr < (unsigned)PSIZE) {
                const float* prow = patch + ch * (PSIZE * PSIZE) + pr * PSIZE;
                float* vf = &v.x;
#pragma unroll
                for (int j = 0; j < 4; ++j) {
                    const int pc = col + j - c;
                    if ((unsigned)pc < (unsigned)PSIZE) vf[j] += prow[pc];
                }
            }
            float* o = out + imgBase + e;
            __builtin_nontemporal_store(v.x, o + 0);
            __builtin_nontemporal_store(v.y, o + 1);
            __builtin_nontemporal_store(v.z, o + 2);
            __builtin_nontemporal_store(v.w, o + 3);
        }
    }
}

extern "C" void kernel_launch(void* const* d_in, const int* in_sizes, int n_in,
                              void* d_out, int out_size, void* d_ws, size_t ws_size,
                              hipStream_t stream) {
    const float* x       = (const float*)d_in[0];
    const float* patch   = (const float*)d_in[1];
    const int*   offsets = (const int*)d_in[2];
    float*       out     = (float*)d_out;

    const int batch = in_sizes[0] / IMG_ELEMS;   // 256 for the reference shapes
    dim3 grid(BLKS_PER_IMG, batch, 1);
    RandomPrompter_kernel<<<grid, TPB, 0, stream>>>(x, patch, offsets, out);
}